// MappedMaxUnpool_34282428956677
// MI455X (gfx1250) — compile-verified
//
#include <hip/hip_runtime.h>
#include <cstdint>

// Problem constants (from reference)
#define B_      8
#define C_      64
#define N_OUT_  65536
#define N_IN_   262144
#define K_      9
#define P_      4

// Tiling
#define TILE_    256   // n_out values staged per block
#define BCPB_    64    // (b,c) planes processed per block (reuse of staged table)
#define THREADS_ 256

typedef float v4f __attribute__((ext_vector_type(4)));

// ---------------------------------------------------------------------------
// CDNA5 async global -> LDS copy (16 bytes per lane). The LDS destination is
// the per-lane 32-bit LDS byte address (low 32 bits of the generic pointer),
// the source is a 64-bit global address in a VGPR pair.
// Tracked by ASYNCcnt; wave waits with s_wait_asynccnt.
// ---------------------------------------------------------------------------
__device__ __forceinline__ void async_g2l_b128(uint32_t lds_addr, const void* gptr) {
  asm volatile("global_load_async_to_lds_b128 %0, %1, off"
               :: "v"(lds_addr), "v"((unsigned long long)(uintptr_t)gptr)
               : "memory");
}

__device__ __forceinline__ void wait_async_zero() {
  asm volatile("s_wait_asynccnt 0" ::: "memory");
}

// ---------------------------------------------------------------------------
// Zero-init of the 537 MB output with NT 128-bit stores (written once more by
// atomics afterwards; keep it out of the near caches).
// ---------------------------------------------------------------------------
__global__ void __launch_bounds__(THREADS_)
zero_f4_kernel(v4f* __restrict__ o, int n4) {
  int i = blockIdx.x * THREADS_ + threadIdx.x;
  if (i < n4) {
    v4f z = {0.f, 0.f, 0.f, 0.f};
    __builtin_nontemporal_store(z, o + i);
  }
}

// ---------------------------------------------------------------------------
// Scatter-add unpool.
//  grid.x = N_OUT/TILE_  (which n_out tile)
//  grid.y = (B*C)/BCPB_  (which group of (b,c) planes)
// Each block:
//  1) async-stages sample_map[n0:n0+256, :, :] and interp_weights[...] into
//     LDS (two contiguous 36 KB regions, 16 B per lane per async op).
//  2) loops over 64 (b,c) planes; per element: NT-load x/idx (streamed once),
//     ds_load_b128 the selected 4-index row + 4-weight row from LDS, and do
//     4 native global_atomic_add_f32 into the output plane.
// ---------------------------------------------------------------------------
__global__ void __launch_bounds__(THREADS_)
mapped_unpool_scatter(const float* __restrict__ x,
                      const int*   __restrict__ idx_mask,
                      const int*   __restrict__ smap,
                      const float* __restrict__ wgt,
                      float*       __restrict__ out) {
  __shared__ __align__(16) int   s_map[TILE_ * K_ * P_];   // 36864 B
  __shared__ __align__(16) float s_wgt[TILE_ * K_ * P_];   // 36864 B

  const int  tid = threadIdx.x;
  const int  n0  = blockIdx.x * TILE_;
  const long tbl = (long)n0 * (K_ * P_);   // element offset of this tile's rows

  // --- Stage tables into LDS (contiguous: tile rows are back-to-back) ------
  const char* gm = (const char*)(smap + tbl);
  const char* gw = (const char*)(wgt  + tbl);
  const uint32_t lm = (uint32_t)(uintptr_t)&s_map[0];
  const uint32_t lw = (uint32_t)(uintptr_t)&s_wgt[0];

  // 36864 bytes / (256 lanes * 16 B) = 9 iterations per table
  #pragma unroll
  for (int i = 0; i < (TILE_ * K_ * P_ * 4) / (THREADS_ * 16); ++i) {
    const int off = (i * THREADS_ + tid) * 16;
    async_g2l_b128(lm + off, gm + off);
    async_g2l_b128(lw + off, gw + off);
  }
  wait_async_zero();
  __syncthreads();

  // --- Main scatter loop ---------------------------------------------------
  const int n   = n0 + tid;                // this thread's n_out (fixed)
  const int bc0 = blockIdx.y * BCPB_;

  for (int j = 0; j < BCPB_; ++j) {
    const int  bc = bc0 + j;
    const long s  = (long)bc * N_OUT_ + n;

    // streamed exactly once across the whole kernel -> non-temporal
    const float xv = __builtin_nontemporal_load(x + s);
    const int   k  = __builtin_nontemporal_load(idx_mask + s);

    // selected row: 16-byte aligned in LDS -> ds_load_b128
    const int r = (tid * K_ + k) * P_;
    const int4   m = *reinterpret_cast<const int4*>(&s_map[r]);
    const float4 w = *reinterpret_cast<const float4*>(&s_wgt[r]);

    float* op = out + (long)bc * N_IN_;
    // native no-return FP32 atomics (global_atomic_add_f32), resolved in L2
    unsafeAtomicAdd(op + m.x, w.x * xv);
    unsafeAtomicAdd(op + m.y, w.y * xv);
    unsafeAtomicAdd(op + m.z, w.z * xv);
    unsafeAtomicAdd(op + m.w, w.w * xv);
  }
}

// ---------------------------------------------------------------------------
extern "C" void kernel_launch(void* const* d_in, const int* in_sizes, int n_in,
                              void* d_out, int out_size, void* d_ws, size_t ws_size,
                              hipStream_t stream) {
  (void)in_sizes; (void)n_in; (void)d_ws; (void)ws_size; (void)out_size;

  const float* x    = (const float*)d_in[0];  // [B,C,N_OUT] f32
  const int*   idx  = (const int*)  d_in[1];  // [B,C,N_OUT] i32
  const int*   smap = (const int*)  d_in[2];  // [N_OUT,K,P] i32
  const float* wgt  = (const float*)d_in[3];  // [N_OUT,K,P] f32
  float*       out  = (float*)d_out;          // [B,C,N_IN]  f32

  // 1) zero the output (harness poisons it)
  const long total = (long)B_ * C_ * N_IN_;   // 134,217,728 floats
  const int  n4    = (int)(total / 4);        // 33,554,432 float4s
  zero_f4_kernel<<<dim3((n4 + THREADS_ - 1) / THREADS_), dim3(THREADS_), 0, stream>>>(
      (v4f*)out, n4);

  // 2) scatter-add unpool
  dim3 grid(N_OUT_ / TILE_, (B_ * C_) / BCPB_);   // 256 x 8
  mapped_unpool_scatter<<<grid, dim3(THREADS_), 0, stream>>>(x, idx, smap, wgt, out);
}